// PointGPTEncoder_10282151706644
// MI455X (gfx1250) — compile-verified
//
#include <hip/hip_runtime.h>

#define BB 16
#define NN 32768
#define GG 128
#define KK 64
#define EDIM 384
#define BNEPS 1e-5f
#define FLT_BIG 3.4e38f

typedef __attribute__((ext_vector_type(2))) float v2f;
typedef __attribute__((ext_vector_type(8))) float v8f;

// ---------------------------------------------------------------------------
// One 16x16 f32 output tile of C = A(MxK, global, row-major lda) * B(KxN=64,
// LDS, row-major ld=64), accumulated K in steps of 4 via V_WMMA_F32_16X16X4_F32.
// A frag: lanes 0-15 M=0..15 K={0,1}; lanes 16-31 M=0..15 K={2,3}.
// B frag: lanes 0-15 N=0..15 K={0,1}; lanes 16-31 N=0..15 K={2,3}.
// C/D   : VGPR v -> row v + (lane>=16)*8, col lane&15.
// ---------------------------------------------------------------------------
__device__ __forceinline__ v8f wmma_tile_f32(const float* __restrict__ A, int lda,
                                             const float* __restrict__ Blds,
                                             int mt, int nt, int Kdim, int lane) {
  v8f c = {0.f, 0.f, 0.f, 0.f, 0.f, 0.f, 0.f, 0.f};
  const int m    = mt * 16 + (lane & 15);
  const int n    = nt * 16 + (lane & 15);
  const int ksub = (lane >> 4) << 1;   // 0 or 2
  const float* arow = A + (size_t)m * lda;
  const float* bcol = Blds + n;
#pragma unroll 4
  for (int k0 = 0; k0 < Kdim; k0 += 4) {
    const int kb = k0 + ksub;
    v2f a = *(const v2f*)(arow + kb);            // lda even, kb even -> 8B aligned
    v2f b;
    b.x = bcol[(size_t)kb * 64];
    b.y = bcol[(size_t)(kb + 1) * 64];
    c = __builtin_amdgcn_wmma_f32_16x16x4_f32(false, a, false, b, (short)0, c,
                                              false, false);
  }
  return c;
}

// ---------------------------------------------------------------------------
// 1) Farthest point sampling. grid = B, block = 1024. dist array in LDS.
//    Matches jax scan semantics: cid[s] = far at start of step s, far0 = 0,
//    argmax first-index tie-break.
// ---------------------------------------------------------------------------
__global__ __launch_bounds__(1024) void fps_kernel(const float* __restrict__ points,
                                                   float* __restrict__ cent) {
  extern __shared__ float sm[];
  float* dist = sm;                 // NN
  float* rd   = sm + NN;            // 1024
  int*   ri   = (int*)(rd + 1024);  // 1024
  __shared__ int s_far;

  const int b   = blockIdx.x;
  const int tid = threadIdx.x;
  const float* pb = points + (size_t)b * NN * 3;

  for (int i = tid; i < NN; i += 1024) dist[i] = FLT_BIG;
  if (tid == 0) s_far = 0;
  __syncthreads();

  for (int s = 0; s < GG; ++s) {
    const int far = s_far;
    const float cx = pb[far * 3 + 0];
    const float cy = pb[far * 3 + 1];
    const float cz = pb[far * 3 + 2];
    if (tid == 0) {
      cent[((size_t)b * GG + s) * 3 + 0] = cx;
      cent[((size_t)b * GG + s) * 3 + 1] = cy;
      cent[((size_t)b * GG + s) * 3 + 2] = cz;
    }
    float best = -FLT_BIG;
    int   bi   = 0x7fffffff;
    for (int i = tid; i < NN; i += 1024) {
      const float dx = pb[i * 3 + 0] - cx;
      const float dy = pb[i * 3 + 1] - cy;
      const float dz = pb[i * 3 + 2] - cz;
      float d = dx * dx + dy * dy + dz * dz;
      float m = fminf(dist[i], d);
      dist[i] = m;
      if (m > best) { best = m; bi = i; }
    }
    rd[tid] = best; ri[tid] = bi;
    __syncthreads();
    for (int off = 512; off > 0; off >>= 1) {
      if (tid < off) {
        float d2 = rd[tid + off]; int i2 = ri[tid + off];
        if (d2 > rd[tid] || (d2 == rd[tid] && i2 < ri[tid])) { rd[tid] = d2; ri[tid] = i2; }
      }
      __syncthreads();
    }
    if (tid == 0) s_far = ri[0];
    __syncthreads();
  }
}

// ---------------------------------------------------------------------------
// 2) kNN (top-64 nearest of 32768) + local coords. grid = B*G, block = 256.
//    Per-thread top-64 of its 128 strided candidates, then 64 global argmin
//    extractions from LDS.
// ---------------------------------------------------------------------------
__global__ __launch_bounds__(256) void knn_kernel(const float* __restrict__ points,
                                                  const float* __restrict__ cent,
                                                  float* __restrict__ localbuf) {
  extern __shared__ float sm[];
  float* cd = sm;                        // 256*64
  int*   ci = (int*)(cd + 256 * 64);     // 256*64
  float* rd = (float*)(ci + 256 * 64);   // 256
  int*   ri = (int*)(rd + 256);          // 256

  const int bg  = blockIdx.x;
  const int b   = bg / GG;
  const int tid = threadIdx.x;
  const float* pb = points + (size_t)b * NN * 3;
  const float cx = cent[(size_t)bg * 3 + 0];
  const float cy = cent[(size_t)bg * 3 + 1];
  const float cz = cent[(size_t)bg * 3 + 2];

  float ld[KK]; int li[KK];
#pragma unroll
  for (int j = 0; j < KK; ++j) { ld[j] = FLT_BIG; li[j] = 0; }
  float worst = FLT_BIG; int wpos = 0;

  for (int i = tid; i < NN; i += 256) {
    const float dx = pb[i * 3 + 0] - cx;
    const float dy = pb[i * 3 + 1] - cy;
    const float dz = pb[i * 3 + 2] - cz;
    const float d = dx * dx + dy * dy + dz * dz;
    if (d < worst) {
      ld[wpos] = d; li[wpos] = i;
      worst = ld[0]; wpos = 0;
      for (int j = 1; j < KK; ++j)
        if (ld[j] > worst) { worst = ld[j]; wpos = j; }
    }
  }
  for (int j = 0; j < KK; ++j) { cd[tid * KK + j] = ld[j]; ci[tid * KK + j] = li[j]; }
  __syncthreads();

  for (int r = 0; r < KK; ++r) {
    float best = FLT_BIG; int bslot = tid * KK;
    for (int j = 0; j < KK; ++j) {
      const float v = cd[tid * KK + j];
      if (v < best) { best = v; bslot = tid * KK + j; }
    }
    rd[tid] = best; ri[tid] = bslot;
    __syncthreads();
    for (int off = 128; off > 0; off >>= 1) {
      if (tid < off) {
        if (rd[tid + off] < rd[tid]) { rd[tid] = rd[tid + off]; ri[tid] = ri[tid + off]; }
      }
      __syncthreads();
    }
    if (tid == 0) {
      const int slot = ri[0];
      const int pi   = ci[slot];
      cd[slot] = FLT_BIG;
      const size_t o = ((size_t)bg * KK + r) * 3;
      localbuf[o + 0] = pb[pi * 3 + 0] - cx;
      localbuf[o + 1] = pb[pi * 3 + 1] - cy;
      localbuf[o + 2] = pb[pi * 3 + 2] - cz;
    }
    __syncthreads();
  }
}

// ---------------------------------------------------------------------------
// 3) Greedy chain ordering (exact reference semantics, incl. row-set-to-inf).
//    grid = B, block = 128. Writes sorted centroids to d_out and perm to ws.
// ---------------------------------------------------------------------------
__global__ __launch_bounds__(128) void morton_kernel(const float* __restrict__ centIn,
                                                     float* __restrict__ centOut,
                                                     int* __restrict__ perm) {
  extern __shared__ float sm[];
  float* C  = sm;               // GG*3
  float* D  = C + GG * 3;       // GG*GG
  float* rv = D + GG * GG;      // GG
  int*   ri = (int*)(rv + GG);  // GG
  int*   order = ri + GG;       // GG
  __shared__ int s_cur;

  const int b = blockIdx.x;
  const int tid = threadIdx.x;
  for (int i = tid; i < GG * 3; i += 128) C[i] = centIn[(size_t)b * GG * 3 + i];
  __syncthreads();
  for (int e = tid; e < GG * GG; e += 128) {
    const int i = e >> 7, j = e & 127;
    if (i == j) D[e] = FLT_BIG;
    else {
      const float dx = C[i * 3 + 0] - C[j * 3 + 0];
      const float dy = C[i * 3 + 1] - C[j * 3 + 1];
      const float dz = C[i * 3 + 2] - C[j * 3 + 2];
      D[e] = dx * dx + dy * dy + dz * dz;
    }
  }
  if (tid == 0) { s_cur = 0; order[0] = 0; }
  __syncthreads();

  for (int step = 1; step < GG; ++step) {
    const int cur = s_cur;
    rv[tid] = D[cur * GG + tid]; ri[tid] = tid;
    __syncthreads();
    for (int off = 64; off > 0; off >>= 1) {
      if (tid < off) {
        float v2 = rv[tid + off]; int i2 = ri[tid + off];
        if (v2 < rv[tid] || (v2 == rv[tid] && i2 < ri[tid])) { rv[tid] = v2; ri[tid] = i2; }
      }
      __syncthreads();
    }
    const int closest = ri[0];
    D[closest * GG + tid] = FLT_BIG;   // reference: df.at[closest].set(inf) (row)
    if (tid == 0) { order[step] = closest; s_cur = closest; }
    __syncthreads();
  }

  perm[(size_t)b * GG + tid] = order[tid];
  const int src = order[tid];
  centOut[((size_t)b * GG + tid) * 3 + 0] = C[src * 3 + 0];
  centOut[((size_t)b * GG + tid) * 3 + 1] = C[src * 3 + 1];
  centOut[((size_t)b * GG + tid) * 3 + 2] = C[src * 3 + 2];
}

// ---------------------------------------------------------------------------
// 4) Encoder. grid = B*G, block = 256 (8 wave32). All activations LDS-resident
//    (~294 KB of the 320 KB WGP LDS). GEMMs via fp32 WMMA 16x16x4.
// ---------------------------------------------------------------------------
__global__ __launch_bounds__(256) void encoder_kernel(
    const float* __restrict__ localbuf, const int* __restrict__ perm,
    const float* __restrict__ w1, const float* __restrict__ b1,
    const float* __restrict__ bn1_g, const float* __restrict__ bn1_b,
    const float* __restrict__ bn1_m, const float* __restrict__ bn1_v,
    const float* __restrict__ w2, const float* __restrict__ b2,
    const float* __restrict__ w3, const float* __restrict__ b3,
    const float* __restrict__ bn2_g, const float* __restrict__ bn2_b,
    const float* __restrict__ bn2_m, const float* __restrict__ bn2_v,
    const float* __restrict__ w4, const float* __restrict__ b4,
    float* __restrict__ tokens) {
  extern __shared__ float sm[];
  float* H    = sm;               // 512*64  (fg rows 0-255, f2 rows 256-511)
  float* F3   = H + 512 * 64;     // 512*64
  float* F1   = F3 + 512 * 64;    // 128*64
  float* PTS  = F1 + 128 * 64;    // 64*3
  float* bn1s = PTS + KK * 3;     // 128
  float* bn1o = bn1s + 128;       // 128
  float* bn2s = bn1o + 128;       // 512
  float* bn2o = bn2s + 512;       // 512
  float* F4   = H;                // alias: 384*64 <= 512*64, H dead by stage D

  const int bg   = blockIdx.x;
  const int b    = bg / GG;
  const int tid  = threadIdx.x;
  const int lane = tid & 31;
  const int wv   = tid >> 5;

  // gather permuted group + precompute BN scale/shift
  const int gsrc = perm[bg];
  const float* lsrc = localbuf + ((size_t)b * GG + gsrc) * KK * 3;
  for (int i = tid; i < KK * 3; i += 256) PTS[i] = lsrc[i];
  if (tid < 128) {
    const float inv = bn1_g[tid] * rsqrtf(bn1_v[tid] + BNEPS);
    bn1s[tid] = inv;
    bn1o[tid] = bn1_b[tid] - bn1_m[tid] * inv;
  }
  for (int m = tid; m < 512; m += 256) {
    const float inv = bn2_g[m] * rsqrtf(bn2_v[m] + BNEPS);
    bn2s[m] = inv;
    bn2o[m] = bn2_b[m] - bn2_m[m] * inv;
  }
  __syncthreads();

  // Stage A: f1 = relu(bn1(w1 @ x^T + b1))  -> F1[128][64]
  for (int e = tid; e < 128 * 64; e += 256) {
    const int r = e >> 6, n = e & 63;
    float v = w1[r * 3 + 0] * PTS[n * 3 + 0] + w1[r * 3 + 1] * PTS[n * 3 + 1] +
              w1[r * 3 + 2] * PTS[n * 3 + 2] + b1[r];
    v = v * bn1s[r] + bn1o[r];
    F1[e] = fmaxf(v, 0.f);
  }
  __syncthreads();

  // Stage B: f2 = w2(256x128) @ f1 + b2 -> H rows 256..511
  for (int t = wv; t < 16 * 4; t += 8) {
    const int mt = t >> 2, nt = t & 3;
    v8f c = wmma_tile_f32(w2, 128, F1, mt, nt, 128, lane);
    const int ncol = nt * 16 + (lane & 15);
#pragma unroll
    for (int v = 0; v < 8; ++v) {
      const int mrow = mt * 16 + v + ((lane >> 4) << 3);
      H[(256 + mrow) * 64 + ncol] = c[v] + b2[mrow];
    }
  }
  __syncthreads();

  // fg = rowmax(f2), broadcast into H rows 0..255
  {
    const int m = tid;  // 256 threads, 256 rows
    float mx = -FLT_BIG;
    for (int n = 0; n < 64; ++n) mx = fmaxf(mx, H[(256 + m) * 64 + n]);
    for (int n = 0; n < 64; ++n) H[m * 64 + n] = mx;
  }
  __syncthreads();

  // Stage C: f3 = relu(bn2(w3(512x512) @ H + b3)) -> F3
  for (int t = wv; t < 32 * 4; t += 8) {
    const int mt = t >> 2, nt = t & 3;
    v8f c = wmma_tile_f32(w3, 512, H, mt, nt, 512, lane);
    const int ncol = nt * 16 + (lane & 15);
#pragma unroll
    for (int v = 0; v < 8; ++v) {
      const int mrow = mt * 16 + v + ((lane >> 4) << 3);
      float val = (c[v] + b3[mrow]) * bn2s[mrow] + bn2o[mrow];
      F3[mrow * 64 + ncol] = fmaxf(val, 0.f);
    }
  }
  __syncthreads();

  // Stage D: f4 = w4(384x512) @ f3 + b4 -> F4 (aliases H)
  for (int t = wv; t < 24 * 4; t += 8) {
    const int mt = t >> 2, nt = t & 3;
    v8f c = wmma_tile_f32(w4, 512, F3, mt, nt, 512, lane);
    const int ncol = nt * 16 + (lane & 15);
#pragma unroll
    for (int v = 0; v < 8; ++v) {
      const int mrow = mt * 16 + v + ((lane >> 4) << 3);
      F4[mrow * 64 + ncol] = c[v] + b4[mrow];
    }
  }
  __syncthreads();

  // token = rowmax(f4)
  for (int m = tid; m < EDIM; m += 256) {
    float mx = -FLT_BIG;
    for (int n = 0; n < 64; ++n) mx = fmaxf(mx, F4[m * 64 + n]);
    tokens[(size_t)bg * EDIM + m] = mx;
  }
}

// ---------------------------------------------------------------------------
extern "C" void kernel_launch(void* const* d_in, const int* in_sizes, int n_in,
                              void* d_out, int out_size, void* d_ws, size_t ws_size,
                              hipStream_t stream) {
  const float* points = (const float*)d_in[0];
  const float* w1 = (const float*)d_in[1];
  const float* b1 = (const float*)d_in[2];
  const float* bn1_g = (const float*)d_in[3];
  const float* bn1_b = (const float*)d_in[4];
  const float* bn1_m = (const float*)d_in[5];
  const float* bn1_v = (const float*)d_in[6];
  const float* w2 = (const float*)d_in[7];
  const float* b2 = (const float*)d_in[8];
  const float* w3 = (const float*)d_in[9];
  const float* b3 = (const float*)d_in[10];
  const float* bn2_g = (const float*)d_in[11];
  const float* bn2_b = (const float*)d_in[12];
  const float* bn2_m = (const float*)d_in[13];
  const float* bn2_v = (const float*)d_in[14];
  const float* w4 = (const float*)d_in[15];
  const float* b4 = (const float*)d_in[16];

  char* ws = (char*)d_ws;
  float* cent     = (float*)ws;                       // B*G*3 f32 (pre-sort)
  int*   perm     = (int*)(ws + BB * GG * 3 * 4);     // B*G i32
  float* localbuf = (float*)(ws + BB * GG * 3 * 4 + BB * GG * 4);  // B*G*K*3 f32

  float* centOut = (float*)d_out;                 // first B*G*3 floats
  float* tokens  = centOut + (size_t)BB * GG * 3; // then B*G*384 floats

  const int fps_shmem = NN * 4 + 1024 * 4 + 1024 * 4;
  fps_kernel<<<BB, 1024, fps_shmem, stream>>>(points, cent);

  const int knn_shmem = 256 * KK * 4 * 2 + 256 * 8;
  knn_kernel<<<BB * GG, 256, knn_shmem, stream>>>(points, cent, localbuf);

  const int mor_shmem = (GG * 3 + GG * GG + GG) * 4 + GG * 4 * 2;
  morton_kernel<<<BB, GG, mor_shmem, stream>>>(cent, centOut, perm);

  const int enc_shmem = (512 * 64 + 512 * 64 + 128 * 64 + KK * 3 + 128 * 2 + 512 * 2) * 4;
  encoder_kernel<<<BB * GG, 256, enc_shmem, stream>>>(
      localbuf, perm, w1, b1, bn1_g, bn1_b, bn1_m, bn1_v, w2, b2, w3, b3,
      bn2_g, bn2_b, bn2_m, bn2_v, w4, b4, tokens);
}